// RBF_24730421691122
// MI455X (gfx1250) — compile-verified
//
#include <hip/hip_runtime.h>
#include <math.h>

#define N_PART 256
#define DIM    65536
#define KSPLIT 32
#define KPER   (DIM / KSPLIT)   // 2048 K elements per grid.z slice
#define KC     32               // K chunk staged in LDS per iteration
#define NCHUNK (KPER / KC)      // 64 double-buffered chunks per slice

typedef __attribute__((ext_vector_type(2))) float v2f;
typedef __attribute__((ext_vector_type(8))) float v8f;

#if defined(__has_builtin)
#if __has_builtin(__builtin_amdgcn_global_load_async_to_lds_b32) && \
    __has_builtin(__builtin_amdgcn_s_wait_asynccnt)
#define HAVE_ASYNC_LDS 1
#endif
#endif

#ifdef HAVE_ASYNC_LDS
// Builtin signature (from hipcc diagnostic): global int* src, LDS int* dst,
// imm offset, imm cpol. Non-const, int-typed, address-space qualified.
typedef __attribute__((address_space(1))) int* gas_ptr;
typedef __attribute__((address_space(3))) int* lds_ptr;
#endif

// ---------------------------------------------------------------------------
// sq[i] = sum_d X[i,d]^2   (one block per row, float4 coalesced loads)
// ---------------------------------------------------------------------------
__global__ void sq_kernel(const float* __restrict__ X, float* __restrict__ sq) {
  __shared__ float red[256];
  const int i = blockIdx.x;
  const int t = threadIdx.x;
  const float4* row = (const float4*)(X + (size_t)i * DIM);
  float acc = 0.0f;
  for (int d4 = t; d4 < DIM / 4; d4 += 256) {
    float4 v = row[d4];
    acc += v.x * v.x + v.y * v.y + v.z * v.z + v.w * v.w;
  }
  red[t] = acc;
  __syncthreads();
  for (int s = 128; s > 0; s >>= 1) {
    if (t < s) red[t] += red[t + s];
    __syncthreads();
  }
  if (t == 0) sq[i] = red[0];
}

// ---------------------------------------------------------------------------
// Partial Gram matrix: part[z][i,j] = sum_{k in slice z} X[i,k]*X[j,k]
// One wave per 16x16 tile; double-buffered async global->LDS staging
// (ASYNCcnt path), WMMA f32 16x16x4 compute.
// ---------------------------------------------------------------------------
__global__ void gram_partial_kernel(const float* __restrict__ X,
                                    float* __restrict__ part) {
  __shared__ float As[2][16][KC + 1];
  __shared__ float Bs[2][16][KC + 1];
  const int lane = threadIdx.x;       // 0..31
  const int half = lane >> 4;         // 0 or 1
  const int l    = lane & 15;
  const int i0   = blockIdx.y * 16;
  const int j0   = blockIdx.x * 16;
  const int kbase = blockIdx.z * KPER;

  // Stage one 16x32 panel pair into buffer `buf` starting at column k0.
  auto stage = [&](int buf, int k0) {
    for (int r = 0; r < 16; ++r) {
      const float* ga = X + (size_t)(i0 + r) * DIM + k0 + lane;
      const float* gb = X + (size_t)(j0 + r) * DIM + k0 + lane;
#ifdef HAVE_ASYNC_LDS
      __builtin_amdgcn_global_load_async_to_lds_b32(
          (gas_ptr)(unsigned long long)ga,
          (lds_ptr)(unsigned long long)&As[buf][r][lane], 0, 0);
      __builtin_amdgcn_global_load_async_to_lds_b32(
          (gas_ptr)(unsigned long long)gb,
          (lds_ptr)(unsigned long long)&Bs[buf][r][lane], 0, 0);
#else
      As[buf][r][lane] = *ga;
      Bs[buf][r][lane] = *gb;
#endif
    }
  };

  v8f c = {};
  stage(0, kbase);
#ifdef HAVE_ASYNC_LDS
  __builtin_amdgcn_s_wait_asynccnt(0);
#endif
  __syncthreads();

  for (int chunk = 0; chunk < NCHUNK; ++chunk) {
    const int cur = chunk & 1;
    if (chunk + 1 < NCHUNK) stage(cur ^ 1, kbase + (chunk + 1) * KC);
#pragma unroll
    for (int kk = 0; kk < KC; kk += 4) {
      v2f a, b;
      // A 16x4: lanes 0-15 carry K=0,1 ; lanes 16-31 carry K=2,3 ; M = l
      a.x = As[cur][l][kk + 2 * half + 0];
      a.y = As[cur][l][kk + 2 * half + 1];
      // B 4x16: B[k][n] = X[j0+n][k]; same half-split over K
      b.x = Bs[cur][l][kk + 2 * half + 0];
      b.y = Bs[cur][l][kk + 2 * half + 1];
      c = __builtin_amdgcn_wmma_f32_16x16x4_f32(false, a, false, b,
                                                (short)0, c, false, false);
    }
#ifdef HAVE_ASYNC_LDS
    __builtin_amdgcn_s_wait_asynccnt(0);
#endif
    __syncthreads();
  }

  // D 16x16 f32: reg r holds M = r + 8*half, N = l
  float* dst = part + (size_t)blockIdx.z * (N_PART * N_PART);
#pragma unroll
  for (int r = 0; r < 8; ++r) {
    int m = r + 8 * half;
    dst[(i0 + m) * N_PART + (j0 + l)] = c[r];
  }
}

// ---------------------------------------------------------------------------
// dists[i,j] = max(0, sq[i] + sq[j] - 2 * sum_z part[z][i,j])
// ---------------------------------------------------------------------------
__global__ void gram_reduce_kernel(const float* __restrict__ part,
                                   const float* __restrict__ sq,
                                   float* __restrict__ dists) {
  const int e = blockIdx.x * blockDim.x + threadIdx.x;   // 0..65535
  const int i = e >> 8;
  const int j = e & 255;
  float s = 0.0f;
  for (int z = 0; z < KSPLIT; ++z)
    s += part[(size_t)z * (N_PART * N_PART) + e];
  float d = sq[i] + sq[j] - 2.0f * s;
  dists[e] = fmaxf(d, 0.0f);
}

// ---------------------------------------------------------------------------
// Median via full bitonic sort of 65536 floats in 256 KB LDS (one WGP),
// then sigma = sqrt(median / (2 ln(N+1))) blended with EMA.
// scal[0] = 1/(2 sigma^2), scal[1] = 1/sigma^2.
// ---------------------------------------------------------------------------
__global__ void median_sigma_kernel(const float* __restrict__ dists,
                                    const float* __restrict__ EMA,
                                    float* __restrict__ sig_out,
                                    float* __restrict__ scal) {
  extern __shared__ float s[];                 // 65536 floats = 256 KB
  const unsigned t = threadIdx.x;              // 0..1023
  const unsigned n = N_PART * N_PART;
  for (unsigned i = t; i < n; i += 1024) s[i] = dists[i];
  __syncthreads();
  for (unsigned k = 2; k <= n; k <<= 1) {
    for (unsigned j = k >> 1; j > 0; j >>= 1) {
      for (unsigned i = t; i < n; i += 1024) {
        unsigned ixj = i ^ j;
        if (ixj > i) {
          float a = s[i], b = s[ixj];
          bool up = ((i & k) == 0);
          if (up ? (a > b) : (a < b)) { s[i] = b; s[ixj] = a; }
        }
      }
      __syncthreads();
    }
  }
  if (t == 0) {
    float med = 0.5f * (s[n / 2 - 1] + s[n / 2]);
    float h = med / (2.0f * logf((float)(N_PART + 1)));
    float sigma = sqrtf(h);
    float e1 = EMA[1];
    sigma = e1 * sigma + (1.0f - e1) * EMA[0];
    sig_out[0] = sigma;
    scal[0] = 1.0f / (2.0f * sigma * sigma);
    scal[1] = 1.0f / (sigma * sigma);
  }
}

// ---------------------------------------------------------------------------
// In-place: k = exp(-dists / (2 sigma^2))
// ---------------------------------------------------------------------------
__global__ void exp_kernel(float* __restrict__ kbuf,
                           const float* __restrict__ scal) {
  const int e = blockIdx.x * blockDim.x + threadIdx.x;
  const float inv2s2 = scal[0];
  kbuf[e] = expf(-kbuf[e] * inv2s2);
}

// ---------------------------------------------------------------------------
// ksum[j] = sum_i k[i,j]   (deterministic, trivial size)
// ---------------------------------------------------------------------------
__global__ void ksum_kernel(const float* __restrict__ kbuf,
                            float* __restrict__ ksum) {
  const int j = threadIdx.x;   // one block of 256
  float s = 0.0f;
  for (int i = 0; i < N_PART; ++i) s += kbuf[i * N_PART + j];
  ksum[j] = s;
}

// ---------------------------------------------------------------------------
// der[j,d] = (sum_i k[i,j] X[i,d] - ksum[j] X[j,d]) / sigma^2
// 8 waves/block share the LDS-staged k-tile (A); B streams from L2-resident X
// with an explicit prefetch of the next K-chunk (global_prefetch_b8).
// ---------------------------------------------------------------------------
__global__ void der_kernel(const float* __restrict__ X,
                           const float* __restrict__ kbuf,
                           const float* __restrict__ ksum,
                           const float* __restrict__ scal,
                           float* __restrict__ der) {
  __shared__ float ks[16][17];          // [K chunk][M]
  const int t    = threadIdx.x;         // 0..255
  const int wave = t >> 5;              // 0..7
  const int lane = t & 31;
  const int half = lane >> 4;
  const int l    = lane & 15;
  const int j0 = blockIdx.y * 16;                // particle (M) tile
  const int d0 = (blockIdx.x * 8 + wave) * 16;   // feature (N) tile
  const float invs2 = scal[1];

  v8f c = {};
  for (int i0 = 0; i0 < N_PART; i0 += 16) {
    // A[m][kk] = k[(i0+kk), j0+m]  staged once per block
    {
      const int kk = t >> 4, m = t & 15;
      ks[kk][m] = kbuf[(i0 + kk) * N_PART + (j0 + m)];
    }
    __syncthreads();
    // Prefetch the next chunk's B rows (two rows per lane-half, one line each)
    if (i0 + 16 < N_PART) {
      __builtin_prefetch(X + (size_t)(i0 + 16 + 2 * half) * DIM + d0 + l, 0, 1);
      __builtin_prefetch(X + (size_t)(i0 + 24 + 2 * half) * DIM + d0 + l, 0, 1);
    }
#pragma unroll
    for (int kk = 0; kk < 16; kk += 4) {
      v2f a, b;
      a.x = ks[kk + 2 * half + 0][l];
      a.y = ks[kk + 2 * half + 1][l];
      b.x = X[(size_t)(i0 + kk + 2 * half + 0) * DIM + d0 + l];
      b.y = X[(size_t)(i0 + kk + 2 * half + 1) * DIM + d0 + l];
      c = __builtin_amdgcn_wmma_f32_16x16x4_f32(false, a, false, b,
                                                (short)0, c, false, false);
    }
    __syncthreads();
  }
#pragma unroll
  for (int r = 0; r < 8; ++r) {
    const int j = j0 + r + 8 * half;
    const int d = d0 + l;
    const float xv = X[(size_t)j * DIM + d];
    der[(size_t)j * DIM + d] = (c[r] - ksum[j] * xv) * invs2;
  }
}

// ---------------------------------------------------------------------------
extern "C" void kernel_launch(void* const* d_in, const int* in_sizes, int n_in,
                              void* d_out, int out_size, void* d_ws, size_t ws_size,
                              hipStream_t stream) {
  (void)in_sizes; (void)n_in; (void)out_size; (void)ws_size;
  const float* X   = (const float*)d_in[0];
  const float* EMA = (const float*)d_in[1];

  float* out  = (float*)d_out;
  float* kbuf = out;                                          // k: 65536
  float* der  = out + N_PART * N_PART;                        // der: 256*65536
  float* sig  = out + N_PART * N_PART + (size_t)N_PART * DIM; // sigma: 1

  float* ws   = (float*)d_ws;
  float* sq   = ws;          // 256 floats
  float* ksum = ws + 256;    // 256 floats
  float* scal = ws + 512;    // 2 floats: 1/(2s^2), 1/s^2

  // Gram partials use the der region (written last) as 8 MB scratch,
  // so we never depend on ws_size beyond ~2 KB.
  float* part = der;         // KSPLIT * 65536 floats

  sq_kernel<<<N_PART, 256, 0, stream>>>(X, sq);
  gram_partial_kernel<<<dim3(16, 16, KSPLIT), 32, 0, stream>>>(X, part);
  gram_reduce_kernel<<<256, 256, 0, stream>>>(part, sq, kbuf);
  median_sigma_kernel<<<1, 1024, (N_PART * N_PART) * sizeof(float), stream>>>(
      kbuf, EMA, sig, scal);
  exp_kernel<<<256, 256, 0, stream>>>(kbuf, scal);
  ksum_kernel<<<1, 256, 0, stream>>>(kbuf, ksum);
  der_kernel<<<dim3(512, 16), 256, 0, stream>>>(X, kbuf, ksum, scal, der);
}